// MIHistogramPenalty_8650064134500
// MI455X (gfx1250) — compile-verified
//
#include <hip/hip_runtime.h>
#include <math.h>

#define KBINS 2048
#define KBITS 11
#define MI_EPS 1e-10f

typedef __attribute__((ext_vector_type(2))) float v2f;
typedef __attribute__((ext_vector_type(8))) float v8f;

// ---------------------------------------------------------------------------
// Wave32 sum of one float per lane using V_WMMA_F32_16X16X4_F32.
// A (16x4 f32): lane l<16 holds A[l][0]=x, A[l][1]=0; lane l>=16 holds
// A[l-16][2]=x, A[l-16][3]=0.  With B = ones(4x16):
//   D[m][n] = x_m + x_{m+16}  (row sums, replicated across n).
// D layout: lane n<16, VGPR r -> D[r][n]; lane n+16, VGPR r -> D[r+8][n].
// So sum of a lane's 8 D regs = sum(rows 0..7) on lanes 0..15 and
// sum(rows 8..15) on lanes 16..31; total = t(lane0) + t(lane16).
// Requires EXEC all ones: call from full, convergent waves only.
// ---------------------------------------------------------------------------
__device__ __forceinline__ float wave_sum_wmma(float x) {
  v2f A; A[0] = x;    A[1] = 0.0f;
  v2f B; B[0] = 1.0f; B[1] = 1.0f;
  v8f C = {};
  v8f D = __builtin_amdgcn_wmma_f32_16x16x4_f32(
      /*neg_a=*/false, A, /*neg_b=*/false, B,
      /*c_mod=*/(short)0, C, /*reuse_a=*/false, /*reuse_b=*/false);
  float t = D[0] + D[1] + D[2] + D[3] + D[4] + D[5] + D[6] + D[7];
  float lo = __int_as_float(__builtin_amdgcn_readlane(__float_as_int(t), 0));
  float hi = __int_as_float(__builtin_amdgcn_readlane(__float_as_int(t), 16));
  return lo + hi;
}

// Block-level sum; valid on threadIdx.x == 0. blockDim.x must be a multiple
// of 32 and all threads must reach this call (no divergence).
__device__ __forceinline__ float block_sum(float x, float* smem) {
  float w = wave_sum_wmma(x);
  int lane = threadIdx.x & 31;
  int wid  = threadIdx.x >> 5;
  if (lane == 0) smem[wid] = w;
  __syncthreads();
  float s = 0.0f;
  if (threadIdx.x == 0) {
    int nw = blockDim.x >> 5;
    for (int i = 0; i < nw; ++i) s += smem[i];
  }
  __syncthreads();
  return s;
}

// ---------------------------------------------------------------------------
// Vectorized zero: n16 = number of 16-byte packets (global_store_b128).
__global__ void k_zero4(uint4* __restrict__ p, size_t n16) {
  size_t i = (size_t)blockIdx.x * blockDim.x + threadIdx.x;
  size_t stride = (size_t)gridDim.x * blockDim.x;
  uint4 z = make_uint4(0u, 0u, 0u, 0u);
  for (; i < n16; i += stride) p[i] = z;
}

// Per-dim marginal histograms: marg[d][v], d in [0,8), v in [0,2048)
__global__ void k_hist_marg(const int* __restrict__ inp,
                            unsigned* __restrict__ marg, int P) {
  int p = blockIdx.x * blockDim.x + threadIdx.x;
  if (p >= P) return;
  const int4* row = (const int4*)(inp + (size_t)p * 8);
  int4 a = row[0];
  int4 b = row[1];
  atomicAdd(&marg[0 * KBINS + a.x], 1u);
  atomicAdd(&marg[1 * KBINS + a.y], 1u);
  atomicAdd(&marg[2 * KBINS + a.z], 1u);
  atomicAdd(&marg[3 * KBINS + a.w], 1u);
  atomicAdd(&marg[4 * KBINS + b.x], 1u);
  atomicAdd(&marg[5 * KBINS + b.y], 1u);
  atomicAdd(&marg[6 * KBINS + b.z], 1u);
  atomicAdd(&marg[7 * KBINS + b.w], 1u);
}

__device__ __forceinline__ int sel8(int d, int4 a, int4 b) {
  int r = a.x;
  r = (d == 1) ? a.y : r;
  r = (d == 2) ? a.z : r;
  r = (d == 3) ? a.w : r;
  r = (d == 4) ? b.x : r;
  r = (d == 5) ? b.y : r;
  r = (d == 6) ? b.z : r;
  r = (d == 7) ? b.w : r;
  return r;
}

// Joint histograms for tuples [tBase, tBase+tCount):
// joint[t - tBase][ v[d1] + (v[d0] << 11) ] += 1
__global__ void k_hist_joint(const int* __restrict__ inp,
                             const int* __restrict__ tdims,
                             unsigned* __restrict__ joint,
                             int P, int tBase, int tCount) {
  int p = blockIdx.x * blockDim.x + threadIdx.x;
  if (p >= P) return;
  const int4* row = (const int4*)(inp + (size_t)p * 8);
  int4 a = row[0];
  int4 b = row[1];
  const size_t nb = (size_t)KBINS * KBINS;
  for (int t = 0; t < tCount; ++t) {
    int d0 = tdims[2 * (tBase + t) + 0];
    int d1 = tdims[2 * (tBase + t) + 1];
    unsigned x0 = (unsigned)sel8(d0, a, b);
    unsigned x1 = (unsigned)sel8(d1, a, b);
    unsigned idx = x1 + (x0 << KBITS);
    atomicAdd(&joint[(size_t)t * nb + idx], 1u);
  }
}

// Entropy of one joint histogram per blockIdx.y; accumulates into Hout[y].
// Reads with b128 packets; fast-skips all-zero packets.
__global__ void k_entropy_joint(const uint4* __restrict__ joint,
                                float* __restrict__ Hout, float invP) {
  __shared__ float smem[8];
  const size_t nq = ((size_t)KBINS * KBINS) >> 2;  // 16B packets per tuple
  const uint4* jt = joint + (size_t)blockIdx.y * nq;
  float local = 0.0f;
  size_t stride = (size_t)gridDim.x * blockDim.x;
  for (size_t i = (size_t)blockIdx.x * blockDim.x + threadIdx.x; i < nq;
       i += stride) {
    uint4 q = jt[i];
    if ((q.x | q.y | q.z | q.w) != 0u) {
      if (q.x) { float pb = (float)q.x * invP; local += pb * log2f(pb + MI_EPS); }
      if (q.y) { float pb = (float)q.y * invP; local += pb * log2f(pb + MI_EPS); }
      if (q.z) { float pb = (float)q.z * invP; local += pb * log2f(pb + MI_EPS); }
      if (q.w) { float pb = (float)q.w * invP; local += pb * log2f(pb + MI_EPS); }
    }
  }
  float s = block_sum(local, smem);
  if (threadIdx.x == 0) atomicAdd(&Hout[blockIdx.y], -s);
}

// Entropy per dim (one block per dim, 8 blocks total).
__global__ void k_entropy_marg(const unsigned* __restrict__ marg,
                               float* __restrict__ Hdim, float invP) {
  __shared__ float smem[8];
  const unsigned* m = marg + (size_t)blockIdx.x * KBINS;
  float local = 0.0f;
  for (int i = threadIdx.x; i < KBINS; i += blockDim.x) {
    unsigned c = m[i];
    if (c) {
      float pb = (float)c * invP;
      local += pb * log2f(pb + MI_EPS);
    }
  }
  float s = block_sum(local, smem);
  if (threadIdx.x == 0) Hdim[blockIdx.x] = -s;
}

// Combine: mi[t] = (Hm - Hj)/Hm with Hm = Hdim[d0] + Hdim[d1].
// out = { mean(mi), mean(Hm), mean(Hj) }
__global__ void k_final(const int* __restrict__ tdims,
                        const float* __restrict__ Hdim,
                        const float* __restrict__ Hjoint,
                        float* __restrict__ out, int T) {
  if (threadIdx.x != 0 || blockIdx.x != 0) return;
  float smi = 0.0f, shm = 0.0f, shj = 0.0f;
  for (int t = 0; t < T; ++t) {
    float hm = Hdim[tdims[2 * t]] + Hdim[tdims[2 * t + 1]];
    float hj = Hjoint[t];
    smi += (hm - hj) / hm;
    shm += hm;
    shj += hj;
  }
  float invT = 1.0f / (float)T;
  out[0] = smi * invT;
  out[1] = shm * invT;
  out[2] = shj * invT;
}

// ---------------------------------------------------------------------------
extern "C" void kernel_launch(void* const* d_in, const int* in_sizes, int n_in,
                              void* d_out, int out_size, void* d_ws,
                              size_t ws_size, hipStream_t stream) {
  const int* inp   = (const int*)d_in[0];   // (P, 8) int32
  const int* tdims = (const int*)d_in[1];   // (T, 2) int32
  float* out = (float*)d_out;               // 3 floats

  int P = in_sizes[0] / 8;
  int T = in_sizes[1] / 2;
  float invP = 1.0f / (float)P;

  // Workspace layout:
  //   [0,   32)      float Hdim[8]
  //   [32, 160)      float Hjoint[32] (T <= 32 used)
  //   [256, 256+64K) unsigned marg[8][2048]
  //   [fixed, ...)   unsigned joint[chunk][K*K]   (16 MiB per tuple)
  char* ws = (char*)d_ws;
  float*    Hdim   = (float*)ws;
  float*    Hjoint = (float*)(ws + 32);
  unsigned* marg   = (unsigned*)(ws + 256);
  const size_t fixed = 256 + (size_t)8 * KBINS * 4;        // 65,792 B (16B mult)
  const size_t perT  = (size_t)KBINS * KBINS * 4;          // 16 MiB
  unsigned* joint = (unsigned*)(ws + fixed);

  int chunk = 1;
  if (ws_size > fixed + perT) chunk = (int)((ws_size - fixed) / perT);
  if (chunk > T) chunk = T;
  if (chunk < 1) chunk = 1;

  const int pBlocks = (P + 255) / 256;

  // Zero fixed region (H accumulators + marginal histograms).
  {
    size_t n16 = fixed / 16;
    int blocks = (int)((n16 + 255) / 256);
    k_zero4<<<blocks, 256, 0, stream>>>((uint4*)ws, n16);
  }

  // Marginals: histogram + per-dim entropy (once; tuple-independent).
  k_hist_marg<<<pBlocks, 256, 0, stream>>>(inp, marg, P);
  k_entropy_marg<<<8, 256, 0, stream>>>(marg, Hdim, invP);

  // Joint histograms + entropies, chunked by workspace capacity.
  for (int t0 = 0; t0 < T; t0 += chunk) {
    int c = (T - t0 < chunk) ? (T - t0) : chunk;
    size_t n16 = ((size_t)c * KBINS * KBINS) / 4;
    k_zero4<<<4096, 256, 0, stream>>>((uint4*)joint, n16);
    k_hist_joint<<<pBlocks, 256, 0, stream>>>(inp, tdims, joint, P, t0, c);
    dim3 grid(1024, c);
    k_entropy_joint<<<grid, 256, 0, stream>>>((const uint4*)joint,
                                              Hjoint + t0, invP);
  }

  k_final<<<1, 1, 0, stream>>>(tdims, Hdim, Hjoint, out, T);
}